// LocalAttention_3891240370622
// MI455X (gfx1250) — compile-verified
//
#include <hip/hip_runtime.h>
#include <math.h>

// ---------------------------------------------------------------------------
// LocalAttention on MI455X (gfx1250), fp32 WMMA (V_WMMA_F32_16X16X4_F32).
//
// B=4 H=128 W=128 C=64, window 5x5 (25 taps), q/k dim 8.
// Pipeline: [zero fold] -> [WMMA projections q,k,v] ->
//           [per-pixel WMMA attention + softmax + atomic fold scatter] ->
//           [out = gamma*fold + x].
// Workspace (floats): q[N*8] | k[N*8] | v[N*64] | fold[N*64] (36 MB, L2-resident)
//
// Roofline: ~6.6 GFLOP fp32 vs 196 MB mandatory traffic (attn output dominates)
// => bandwidth-bound at ~8.5 us @ 23.3 TB/s once matmuls run on WMMA.
// ---------------------------------------------------------------------------

typedef __attribute__((ext_vector_type(2))) float v2f;
typedef __attribute__((ext_vector_type(8))) float v8f;

#define NB   4
#define NH   128
#define NW   128
#define NC   64
#define NCQ  8
#define NPIX (NB * NH * NW)   // 65536
#define PDIM 25

static __device__ __forceinline__ v8f wmma_f32(v2f a, v2f b, v8f c) {
  // (neg_a, A, neg_b, B, c_mod, C, reuse_a, reuse_b)
  return __builtin_amdgcn_wmma_f32_16x16x4_f32(false, a, false, b, (short)0, c,
                                               false, false);
}

// ---------------------------------------------------------------------------
// Kernel 1: zero the fold accumulator.
// ---------------------------------------------------------------------------
__global__ __launch_bounds__(256) void zero_kernel(float4* __restrict__ p, int n4) {
  int i = blockIdx.x * blockDim.x + threadIdx.x;
  if (i < n4) p[i] = make_float4(0.f, 0.f, 0.f, 0.f);
}

// ---------------------------------------------------------------------------
// Kernel 2: projections.  Per wave: 16 pixels (A rows = NHWC pixel rows).
// qk tile: cols 0-7 = Wq, cols 8-15 = Wk.  v: 4 tiles of 16 cols from Wv.
// 32-bit A 16x4 layout: m=lane&15; lanes<16 hold K0,K1; lanes>=16 hold K2,K3.
// B mirrors with n=lane&15.  D: d[r] = out[m = r+8*(lane>>4)][n = lane&15].
// ---------------------------------------------------------------------------
__global__ __launch_bounds__(256) void proj_kernel(
    const float* __restrict__ x,
    const float* __restrict__ Wq, const float* __restrict__ bq,
    const float* __restrict__ Wk, const float* __restrict__ bk,
    const float* __restrict__ Wv, const float* __restrict__ bv,
    float* __restrict__ qbuf, float* __restrict__ kbuf, float* __restrict__ vbuf) {
  const int lane = threadIdx.x & 31;
  const int wave = threadIdx.x >> 5;
  const int pixBase = (blockIdx.x * 8 + wave) * 16;
  const int half = lane >> 4;
  const int lid = lane & 15;

  v8f accqk = {0.f, 0.f, 0.f, 0.f, 0.f, 0.f, 0.f, 0.f};
  v8f accv[4];
#pragma unroll
  for (int t = 0; t < 4; ++t) accv[t] = accqk;

  const float* wqk = (lid < 8) ? (Wq + lid * NC) : (Wk + (lid - 8) * NC);
  const float* xrow = x + (size_t)(pixBase + lid) * NC;

#pragma unroll
  for (int ks = 0; ks < 16; ++ks) {
    const int k0 = ks * 4 + half * 2;
    v2f a;
    a[0] = xrow[k0];
    a[1] = xrow[k0 + 1];
    v2f bqkf;
    bqkf[0] = wqk[k0];
    bqkf[1] = wqk[k0 + 1];
    accqk = wmma_f32(a, bqkf, accqk);
#pragma unroll
    for (int nt = 0; nt < 4; ++nt) {
      const float* wr = Wv + (nt * 16 + lid) * NC;
      v2f bvf;
      bvf[0] = wr[k0];
      bvf[1] = wr[k0 + 1];
      accv[nt] = wmma_f32(a, bvf, accv[nt]);
    }
  }

  {
    const float bias = (lid < 8) ? bq[lid] : bk[lid - 8];
#pragma unroll
    for (int r = 0; r < 8; ++r) {
      const int pix = pixBase + r + half * 8;
      const float val = accqk[r] + bias;
      if (lid < 8)
        qbuf[pix * NCQ + lid] = val;
      else
        kbuf[pix * NCQ + (lid - 8)] = val;
    }
  }
#pragma unroll
  for (int nt = 0; nt < 4; ++nt) {
    const int n = nt * 16 + lid;
    const float bias = bv[n];
#pragma unroll
    for (int r = 0; r < 8; ++r) {
      const int pix = pixBase + r + half * 8;
      vbuf[pix * NC + n] = accv[nt][r] + bias;
    }
  }
}

// ---------------------------------------------------------------------------
// Kernel 3: per-pixel attention.  One wave per pixel, 4 waves / block.
// Raw-reinterp gathers via per-wave LDS neighbor tables:
//   nbr8[p]  = neighbor_pixel*8  (q/k rows),  -1 if outside image
//   nbr64[p] = neighbor_pixel*64 (v rows & fold targets), -1 if outside
// Q'[m][k]: f=m*8+k -> (c=f/25, p=f%25) -> qbuf[nbr8[p]+c]
// logits 25x25 padded to 32x32 by zeroed fragment lanes -> LDS -> softmax
// (attn drained coalesced from LDS) -> local = S(32x28) x V'(28x64) ->
// fold scatter: f=n*64+cc -> (c=f/25, p=f%25) -> atomicAdd(fold[nbr64[p]+c]).
// ---------------------------------------------------------------------------
__global__ __launch_bounds__(128) void attn_kernel(
    const float* __restrict__ qbuf, const float* __restrict__ kbuf,
    const float* __restrict__ vbuf, float* __restrict__ attn_out,
    float* __restrict__ fold) {
  __shared__ float S[4][32][32];
  __shared__ int NB8[4][32];
  __shared__ int NB64[4][32];
  const int lane = threadIdx.x & 31;
  const int wave = threadIdx.x >> 5;
  const int pid = blockIdx.x * 4 + wave;
  const int b = pid >> 14;
  const int hw = pid & 16383;
  const int h = hw >> 7;
  const int w = hw & 127;
  const int half = lane >> 4;
  const int lid = lane & 15;
  float(*Sw)[32] = S[wave];
  int* nbr8 = NB8[wave];
  int* nbr64 = NB64[wave];

  // ---- per-wave neighbor tables + prefetch of v rows into WGP$ ----
  if (lane < 32) {
    int o8 = -1, o64 = -1;
    if (lane < PDIM) {
      const int hh = h - 2 + lane / 5, ww = w - 2 + lane % 5;
      if ((unsigned)hh < NH && (unsigned)ww < NW) {
        const int base = (b * NH + hh) * NW + ww;
        o8 = base * NCQ;
        o64 = base * NC;
      }
    }
    nbr8[lane] = o8;
    nbr64[lane] = o64;
    if (o64 >= 0) {
      __builtin_prefetch(vbuf + o64, 0, 3);       // global_prefetch_b8
      __builtin_prefetch(vbuf + o64 + 32, 0, 3);  // second cacheline of row
    }
  }
  __syncthreads();

  // ---- logits = Q' K'^T : 2x2 tiles, K=8 (2 k-steps), fragments hoisted ----
  v2f aF[2][2], bF[2][2];
#pragma unroll
  for (int t = 0; t < 2; ++t) {
#pragma unroll
    for (int kbs = 0; kbs < 2; ++kbs) {
      const int m = t * 16 + lid;
#pragma unroll
      for (int r = 0; r < 2; ++r) {
        const int k = kbs * 4 + half * 2 + r;
        float av = 0.f, bvv = 0.f;
        if (m < PDIM) {
          const int f = m * NCQ + k, c = f / 25, p = f - c * 25;
          const int o = nbr8[p];
          if (o >= 0) {
            av = qbuf[o + c];
            bvv = kbuf[o + c];
          }
        }
        aF[t][kbs][r] = av;
        bF[t][kbs][r] = bvv;
      }
    }
  }
#pragma unroll
  for (int mt = 0; mt < 2; ++mt) {
#pragma unroll
    for (int nt = 0; nt < 2; ++nt) {
      v8f acc = {0.f, 0.f, 0.f, 0.f, 0.f, 0.f, 0.f, 0.f};
      acc = wmma_f32(aF[mt][0], bF[nt][0], acc);
      acc = wmma_f32(aF[mt][1], bF[nt][1], acc);
#pragma unroll
      for (int r = 0; r < 8; ++r)
        Sw[mt * 16 + r + half * 8][nt * 16 + lid] = acc[r];
    }
  }
  __syncthreads();

  // ---- softmax over columns, one LDS row per lane (rows 0..24) ----
  if (lane < PDIM) {
    float row[PDIM];
    float mx = -INFINITY;
#pragma unroll
    for (int m = 0; m < PDIM; ++m) {
      row[m] = Sw[lane][m];
      mx = fmaxf(mx, row[m]);
    }
    float s = 0.f;
#pragma unroll
    for (int m = 0; m < PDIM; ++m) {
      row[m] = __expf(row[m] - mx);
      s += row[m];
    }
    const float inv = 1.f / s;
#pragma unroll
    for (int m = 0; m < PDIM; ++m) Sw[lane][m] = row[m] * inv;
#pragma unroll
    for (int m = PDIM; m < 32; ++m) Sw[lane][m] = 0.f;  // K padding for S*V'
  }
  __syncthreads();

  // ---- coalesced attn drain: 625 floats per pixel ----
  {
    float* ao = attn_out + (size_t)pid * (PDIM * PDIM);
#pragma unroll
    for (int t0 = 0; t0 < PDIM * PDIM; t0 += 32) {
      const int t = t0 + lane;
      if (t < PDIM * PDIM) {
        const int n = t / 25, m = t - n * 25;
        ao[t] = Sw[n][m];
      }
    }
  }

  // ---- preload all S fragments (A of local matmul) once ----
  v2f aS[2][7];
#pragma unroll
  for (int mt = 0; mt < 2; ++mt)
#pragma unroll
    for (int kbs = 0; kbs < 7; ++kbs) {
#pragma unroll
      for (int r = 0; r < 2; ++r)
        aS[mt][kbs][r] = Sw[mt * 16 + lid][kbs * 4 + half * 2 + r];
    }

  // ---- local = S x V' : 2x4 tiles, K=28 (7 k-steps), fold scatter ----
#pragma unroll
  for (int nt = 0; nt < 4; ++nt) {
    v8f acc0 = {0.f, 0.f, 0.f, 0.f, 0.f, 0.f, 0.f, 0.f};
    v8f acc1 = acc0;
    const int n = nt * 16 + lid;
#pragma unroll
    for (int kbs = 0; kbs < 7; ++kbs) {
      v2f bb;
#pragma unroll
      for (int r = 0; r < 2; ++r) {
        const int k = kbs * 4 + half * 2 + r;  // V' row (K dim), 0..27
        float bvv = 0.f;
        if (k < PDIM) {
          const int f = k * NC + n, c = f / 25, p = f - c * 25;
          const int o = nbr64[p];
          if (o >= 0) bvv = vbuf[o + c];
        }
        bb[r] = bvv;
      }
      acc0 = wmma_f32(aS[0][kbs], bb, acc0);
      acc1 = wmma_f32(aS[1][kbs], bb, acc1);
    }
#pragma unroll
    for (int r = 0; r < 8; ++r) {
#pragma unroll
      for (int mt = 0; mt < 2; ++mt) {
        const int nrow = mt * 16 + r + half * 8;
        if (nrow < PDIM) {
          const float val = mt ? acc1[r] : acc0[r];
          const int f = nrow * NC + n;
          const int c = f / 25, p = f - c * 25;
          const int o = nbr64[p];
          if (o >= 0) atomicAdd(&fold[o + c], val);
        }
      }
    }
  }
}

// ---------------------------------------------------------------------------
// Kernel 4: out = gamma*fold + x  (NHWC, float4 vectorized)
// ---------------------------------------------------------------------------
__global__ __launch_bounds__(256) void combine_kernel(
    const float4* __restrict__ fold, const float4* __restrict__ x,
    const float* __restrict__ gamma, float4* __restrict__ out, int n4) {
  const int i = blockIdx.x * blockDim.x + threadIdx.x;
  if (i < n4) {
    const float g = gamma[0];
    const float4 f = fold[i];
    const float4 xv = x[i];
    out[i] = make_float4(g * f.x + xv.x, g * f.y + xv.y, g * f.z + xv.z,
                         g * f.w + xv.w);
  }
}

extern "C" void kernel_launch(void* const* d_in, const int* in_sizes, int n_in,
                              void* d_out, int out_size, void* d_ws,
                              size_t ws_size, hipStream_t stream) {
  (void)in_sizes; (void)n_in; (void)out_size; (void)ws_size;
  const float* x  = (const float*)d_in[0];
  const float* Wq = (const float*)d_in[1];
  const float* bq = (const float*)d_in[2];
  const float* Wk = (const float*)d_in[3];
  const float* bk = (const float*)d_in[4];
  const float* Wv = (const float*)d_in[5];
  const float* bv = (const float*)d_in[6];
  const float* gm = (const float*)d_in[7];

  float* out_img = (float*)d_out;                      // [B,H,W,C]
  float* attn    = (float*)d_out + (size_t)NPIX * NC;  // [B,H,W,25,25]

  float* ws   = (float*)d_ws;
  float* qbuf = ws;                              // NPIX*8
  float* kbuf = ws + (size_t)NPIX * NCQ;         // NPIX*8
  float* vbuf = ws + (size_t)NPIX * NCQ * 2;     // NPIX*64
  float* fold = vbuf + (size_t)NPIX * NC;        // NPIX*64

  const int n4 = NPIX * NC / 4;  // 1,048,576 float4
  zero_kernel<<<(n4 + 255) / 256, 256, 0, stream>>>((float4*)fold, n4);
  proj_kernel<<<NPIX / 128, 256, 0, stream>>>(x, Wq, bq, Wk, bk, Wv, bv, qbuf,
                                              kbuf, vbuf);
  attn_kernel<<<NPIX / 4, 128, 0, stream>>>(qbuf, kbuf, vbuf, attn, fold);
  combine_kernel<<<(n4 + 255) / 256, 256, 0, stream>>>(
      (const float4*)fold, (const float4*)x, gm, (float4*)out_img, n4);
}